// RTDETRPostProcessor_15814069584458
// MI455X (gfx1250) — compile-verified
//
#include <hip/hip_runtime.h>
#include <hip/hip_bf16.h>
#include <math.h>

#define B   256
#define Q   1000
#define C   80
#define N   (Q * C)      // 80000 logits per batch
#define NF4 (N / 4)      // 20000 float4 per batch
#define K   300
#define NB  2048         // histogram bins (11 bits of monotonic key)
#define CAP 2048         // candidate capacity (power of 2 for bitonic)
#define TPB 1024
#define TILE 1024        // float4 per async tile (16KB)

// ---- CDNA5 async global->LDS path (guarded; fallback = sync copy) ----
#if defined(__has_builtin)
#if __has_builtin(__builtin_amdgcn_global_load_async_to_lds_b128) && \
    __has_builtin(__builtin_amdgcn_s_wait_asynccnt)
#define USE_ASYNC 1
#endif
#endif

// Builtin signature (from hipcc diagnostic): pointee is GCC-vector int4;
// arg0 = global (AS1) src, arg1 = LDS (AS3) dst, then imm offset, imm cpol.
typedef int __v4i __attribute__((vector_size(16)));
typedef __attribute__((address_space(1))) __v4i __gm_v4i;
typedef __attribute__((address_space(3))) __v4i __lds_v4i;

#ifdef USE_ASYNC
#define ASYNC_LOAD_F4(gp, lp) \
  __builtin_amdgcn_global_load_async_to_lds_b128((__gm_v4i*)(gp), (__lds_v4i*)(lp), 0, 0)
#define WAIT_ASYNC_1() __builtin_amdgcn_s_wait_asynccnt(1)
#define WAIT_ASYNC_0() __builtin_amdgcn_s_wait_asynccnt(0)
#else
#define ASYNC_LOAD_F4(gp, lp) (*(lp) = *(gp))
#define WAIT_ASYNC_1()
#define WAIT_ASYNC_0()
#endif

// Order-preserving float->uint transform (monotonic: bigger float => bigger key)
__device__ __forceinline__ unsigned mono_key(float f) {
  unsigned u = __float_as_uint(f);
  return (u & 0x80000000u) ? ~u : (u | 0x80000000u);
}
__device__ __forceinline__ float inv_mono(unsigned k) {
  unsigned u = (k & 0x80000000u) ? (k ^ 0x80000000u) : ~k;
  return __uint_as_float(u);
}

extern "C" __global__ __launch_bounds__(TPB)
void rtdetr_post_kernel(const float* __restrict__ logits,
                        const float* __restrict__ pboxes,
                        const float* __restrict__ sizes,
                        float* __restrict__ out) {
  const int b = blockIdx.x;
  const unsigned tid = threadIdx.x;
  const float4* gl = (const float4*)(logits + (size_t)b * N);

  // 32KB region: async stage buffers in phase A, candidate array afterwards
  __shared__ __align__(16) unsigned char smem[2 * TILE * sizeof(float4)];
  __shared__ unsigned hist[NB];
  __shared__ unsigned scanb[NB];
  __shared__ unsigned sT, sAbove, sTot, sT2, sRef, sCnt;

  float4 (*stage)[TILE] = reinterpret_cast<float4 (*)[TILE]>(smem);
  unsigned long long* cand = reinterpret_cast<unsigned long long*>(smem);

  for (unsigned i = tid; i < NB; i += TPB) hist[i] = 0u;
  if (tid == 0) { sRef = 0u; sT2 = 0u; sCnt = 0u; }
  __syncthreads();

  // ---------- Phase A: histogram, double-buffered async global->LDS stream ----------
  const int nt = (NF4 + TILE - 1) / TILE;  // 20 tiles
  {
    int j0 = (int)tid;
    if (j0 < NF4) ASYNC_LOAD_F4(gl + j0, &stage[0][tid]);
  }
  for (int t = 0; t < nt; ++t) {
    if (t + 1 < nt) {
      int j = (t + 1) * TILE + (int)tid;
      if (j < NF4) ASYNC_LOAD_F4(gl + j, &stage[(t + 1) & 1][tid]);
      WAIT_ASYNC_1();          // oldest (tile t) complete for this wave
    } else {
      WAIT_ASYNC_0();
    }
    __syncthreads();           // all waves' tile-t data visible in LDS
    int j = t * TILE + (int)tid;
    if (j < NF4) {
      float4 v = stage[t & 1][tid];
      atomicAdd(&hist[mono_key(v.x) >> 21], 1u);
      atomicAdd(&hist[mono_key(v.y) >> 21], 1u);
      atomicAdd(&hist[mono_key(v.z) >> 21], 1u);
      atomicAdd(&hist[mono_key(v.w) >> 21], 1u);
    }
    __syncthreads();           // buffer reuse safe
  }

  // ---------- Phase B: suffix scan, threshold bin ----------
  auto do_scan = [&]() -> const unsigned* {
    unsigned* src = hist;
    unsigned* dst = scanb;
    for (unsigned d = 1; d < NB; d <<= 1) {
      for (unsigned i = tid; i < NB; i += TPB) {
        unsigned v = src[i];
        if (i + d < NB) v += src[i + d];
        dst[i] = v;
      }
      __syncthreads();
      unsigned* tmp = src; src = dst; dst = tmp;
    }
    return src;  // suffix sums: suf[i] = count of keys with bin >= i
  };

  const unsigned* suf = do_scan();
  for (unsigned i = tid; i < NB; i += TPB) {
    unsigned s  = suf[i];
    unsigned sn = (i + 1 < NB) ? suf[i + 1] : 0u;
    if (s >= (unsigned)K && sn < (unsigned)K) { sT = i; sAbove = sn; sTot = s; }
  }
  __syncthreads();
  const unsigned T = sT, above = sAbove, tot = sTot;

  // ---------- optional level-2 refinement (keeps candidate set <= CAP, deterministic) ----------
  if (tot > CAP) {   // uniform across block
    for (unsigned i = tid; i < NB; i += TPB) hist[i] = 0u;
    __syncthreads();
    for (unsigned j = tid; j < NF4; j += TPB) {
      float4 v = gl[j];
      unsigned k0 = mono_key(v.x), k1 = mono_key(v.y);
      unsigned k2 = mono_key(v.z), k3 = mono_key(v.w);
      if ((k0 >> 21) == T) atomicAdd(&hist[(k0 >> 10) & (NB - 1)], 1u);
      if ((k1 >> 21) == T) atomicAdd(&hist[(k1 >> 10) & (NB - 1)], 1u);
      if ((k2 >> 21) == T) atomicAdd(&hist[(k2 >> 10) & (NB - 1)], 1u);
      if ((k3 >> 21) == T) atomicAdd(&hist[(k3 >> 10) & (NB - 1)], 1u);
    }
    __syncthreads();
    const unsigned* suf2 = do_scan();
    unsigned K2 = (unsigned)K - above;
    for (unsigned i = tid; i < NB; i += TPB) {
      unsigned s  = suf2[i];
      unsigned sn = (i + 1 < NB) ? suf2[i + 1] : 0u;
      if (s >= K2 && sn < K2) sT2 = i;
    }
    if (tid == 0) sRef = 1u;
    __syncthreads();
  }
  const unsigned T2 = sT2;
  const unsigned refined = sRef;

  // ---------- Phase C: compact candidates into LDS (L2-resident re-read) ----------
  for (unsigned j = tid; j < NF4; j += TPB) {
    float4 v = gl[j];
    unsigned base = j * 4u;
    float fv[4] = {v.x, v.y, v.z, v.w};
#pragma unroll
    for (int c = 0; c < 4; ++c) {
      unsigned key = mono_key(fv[c]);
      unsigned bin = key >> 21;
      bool sel = (bin > T) ||
                 (bin == T && (!refined || (((key >> 10) & (NB - 1)) >= T2)));
      if (sel) {
        unsigned pos = atomicAdd(&sCnt, 1u);
        if (pos < CAP) {
          // value desc, then index asc  ==  u64 desc on (key , ~idx)
          cand[pos] = ((unsigned long long)key << 32) |
                      (unsigned long long)(0xFFFFFFFFu - (base + (unsigned)c));
        }
      }
    }
  }
  __syncthreads();
  unsigned cnt = sCnt; if (cnt > CAP) cnt = CAP;
  for (unsigned i = cnt + tid; i < CAP; i += TPB) cand[i] = 0ull;  // sinks to bottom
  __syncthreads();

  // ---------- Phase D: bitonic sort, descending, 2048 elems / 1024 threads ----------
  for (unsigned kk = 2; kk <= CAP; kk <<= 1) {
    for (unsigned j = kk >> 1; j > 0; j >>= 1) {
      unsigned idx = ((tid & ~(j - 1)) << 1) | (tid & (j - 1));
      unsigned par = idx | j;
      unsigned long long a = cand[idx];
      unsigned long long c2 = cand[par];
      bool desc = ((idx & kk) == 0);
      if ((a < c2) == desc) { cand[idx] = c2; cand[par] = a; }
      __syncthreads();
    }
  }

  // ---------- Phase E: emit top-300 ----------
  if (tid < K) {
    unsigned long long e = cand[tid];
    unsigned key = (unsigned)(e >> 32);
    unsigned idx = 0xFFFFFFFFu - (unsigned)(e & 0xFFFFFFFFull);
    unsigned lab = idx % C;
    unsigned q   = idx / C;
    float logit  = inv_mono(key);
    float score  = 1.0f / (1.0f + expf(-logit));

    float4 bx = ((const float4*)pboxes)[(size_t)b * Q + q];
    float W = sizes[2 * b], H = sizes[2 * b + 1];
    float4 o;
    o.x = (bx.x - 0.5f * bx.z) * W;
    o.y = (bx.y - 0.5f * bx.w) * H;
    o.z = (bx.x + 0.5f * bx.z) * W;
    o.w = (bx.y + 0.5f * bx.w) * H;

    // output layout: labels [B*K] | boxes [B*K*4] | scores [B*K]
    out[(size_t)b * K + tid] = (float)lab;
    ((float4*)(out + (size_t)B * K))[(size_t)b * K + tid] = o;
    out[(size_t)B * K * 5 + (size_t)b * K + tid] = score;
  }
}

extern "C" void kernel_launch(void* const* d_in, const int* in_sizes, int n_in,
                              void* d_out, int out_size, void* d_ws, size_t ws_size,
                              hipStream_t stream) {
  const float* logits = (const float*)d_in[0];   // [256,1000,80]
  const float* boxes  = (const float*)d_in[1];   // [256,1000,4]
  const float* sizes  = (const float*)d_in[2];   // [256,2]
  float* out = (float*)d_out;                    // 256*300*6 floats
  (void)in_sizes; (void)n_in; (void)out_size; (void)d_ws; (void)ws_size;
  hipLaunchKernelGGL(rtdetr_post_kernel, dim3(B), dim3(TPB), 0, stream,
                     logits, boxes, sizes, out);
}